// CustomResNet_3049426780545
// MI455X (gfx1250) — compile-verified
//
#include <hip/hip_runtime.h>

// Deformable conv (CustomResNet) for gfx1250 / MI455X.
// Stage 1: offset = conv2d(x, offset_weight, stride 2, pad 3)  -> d_ws
// Stage 2: bilinear-gather val tile into LDS, GEMM vs deform_weight -> d_out
// Both GEMMs use V_WMMA_F32_16X16X4_F32 (fp32 in/out, no precision loss).
// Inner loops are branchless: incremental (c,ky,kx) decode instead of
// divisions, clamped-address loads + select instead of EXEC-masked loads.

typedef __attribute__((ext_vector_type(2))) float v2f;
typedef __attribute__((ext_vector_type(8))) float v8f;

#define B_     16
#define C_     3
#define H_     224
#define W_     224
#define O_     98
#define KHW_   49      // 7x7
#define CK_    147     // C_*KHW_
#define CKP_   148     // padded K (multiple of 4)
#define HO_    112
#define WO_    112
#define HOWO_  (HO_*WO_)      // 12544
#define NPIX_  (B_*HOWO_)     // 200704
#define WAVES_PER_BLOCK 4
#define VAL_ROWS   CKP_            // LDS val tile padded to 148 rows
#define VAL_STRIDE (VAL_ROWS*16)   // floats per wave tile in LDS

__device__ __forceinline__ v8f zero8() {
    v8f z;
    #pragma unroll
    for (int i = 0; i < 8; ++i) z[i] = 0.0f;
    return z;
}

// Branchless im2col sample: unconditional clamped load + select.
__device__ __forceinline__ float row_val(const float* __restrict__ xb,
                                         int h2, int w2, int c, int ky, int kx) {
    const int iy = h2 + ky;
    const int ix = w2 + kx;
    const bool valid = (c < C_) & (iy >= 0) & (iy < H_) & (ix >= 0) & (ix < W_);
    const int iyc = min(max(iy, 0), H_ - 1);
    const int ixc = min(max(ix, 0), W_ - 1);
    const int cc  = min(c, C_ - 1);
    const float v = xb[cc * (H_*W_) + iyc * W_ + ixc];
    return valid ? v : 0.0f;
}

// Advance a ck-row decode by 4 rows: r += 4  =>  kx+=4 with carries.
__device__ __forceinline__ void adv4(int& c, int& ky, int& kx) {
    kx += 4;
    if (kx >= 7) { kx -= 7; ky += 1; }   // v_cmp + v_cndmask
    if (ky >= 7) { ky -= 7; c  += 1; }
}

// Branchless weight fragment element: clamped load + select.
__device__ __forceinline__ float wgt_val(const float* __restrict__ wrow,
                                         bool vo, int k) {
    const float t = wrow[min(k, CK_ - 1)];
    return (vo & (k < CK_)) ? t : 0.0f;
}

__device__ __forceinline__ float sample_x(const float* __restrict__ xc,
                                          int iy, int ix) {
    const bool valid = (iy >= 0) & (iy < H_) & (ix >= 0) & (ix < W_);
    const int iyc = min(max(iy, 0), H_ - 1);
    const int ixc = min(max(ix, 0), W_ - 1);
    const float v = xc[iyc * W_ + ixc];
    return valid ? v : 0.0f;
}

// ---------------- Stage 1: offset conv (implicit GEMM, fp32 WMMA) ------------
__global__ __launch_bounds__(32 * WAVES_PER_BLOCK)
void offset_conv_kernel(const float* __restrict__ x,
                        const float* __restrict__ ow,
                        float* __restrict__ off) {
    const int lane = threadIdx.x & 31;
    const int wv   = threadIdx.x >> 5;
    const int p    = lane & 15;      // pixel within tile / N col / M row of A
    const int half = lane >> 4;      // K-pair selector

    const int tile = blockIdx.x * WAVES_PER_BLOCK + wv;   // 12544 tiles
    const int pix  = tile * 16 + p;
    const int b    = pix / HOWO_;
    const int rem  = pix - b * HOWO_;
    const int h    = rem / WO_;
    const int w    = rem - h * WO_;
    const int h2   = h * 2 - 3;
    const int w2   = w * 2 - 3;
    const float* xb = x + b * (C_ * H_ * W_);

    v8f acc[7];
    #pragma unroll
    for (int ot = 0; ot < 7; ++ot) acc[ot] = zero8();

    // Incremental decode of the two ck rows this lane supplies (r0=2*half, r0+1).
    int c0 = 0, ky0 = 0, kx0 = 2 * half;       // row r0 in [0,4)
    int c1 = 0, ky1 = 0, kx1 = 2 * half + 1;

    for (int ks = 0; ks < CKP_; ks += 4) {
        const int k0 = ks + 2 * half;            // K = 2*half + vgpr
        v2f bfrag;
        bfrag.x = row_val(xb, h2, w2, c0, ky0, kx0);
        bfrag.y = row_val(xb, h2, w2, c1, ky1, kx1);
        adv4(c0, ky0, kx0);
        adv4(c1, ky1, kx1);
        #pragma unroll
        for (int ot = 0; ot < 7; ++ot) {
            const int o  = ot * 16 + p;
            const bool vo = (o < O_);
            const float* wrow = ow + min(o, O_ - 1) * CK_;
            v2f afrag;
            afrag.x = wgt_val(wrow, vo, k0);
            afrag.y = wgt_val(wrow, vo, k0 + 1);
            acc[ot] = __builtin_amdgcn_wmma_f32_16x16x4_f32(
                false, afrag, false, bfrag, (short)0, acc[ot], false, false);
        }
    }

    const int hw = h * WO_ + w;
    #pragma unroll
    for (int ot = 0; ot < 7; ++ot) {
        #pragma unroll
        for (int v = 0; v < 8; ++v) {
            const int o = ot * 16 + v + 8 * half;   // D: M = v + 8*half
            if (o < O_) off[(b * O_ + o) * HOWO_ + hw] = acc[ot][v];
        }
    }
}

// ---------------- Stage 2: bilinear gather + deform GEMM ---------------------
__global__ __launch_bounds__(32 * WAVES_PER_BLOCK)
void deform_conv_kernel(const float* __restrict__ x,
                        const float* __restrict__ dw,
                        const float* __restrict__ off,
                        float* __restrict__ out) {
    __shared__ float sval[WAVES_PER_BLOCK * VAL_STRIDE];   // 37.9 KB

    const int lane = threadIdx.x & 31;
    const int wv   = threadIdx.x >> 5;
    const int p    = lane & 15;
    const int half = lane >> 4;

    const int tile = blockIdx.x * WAVES_PER_BLOCK + wv;
    const int pix  = tile * 16 + p;
    const int b    = pix / HOWO_;
    const int rem  = pix - b * HOWO_;
    const int h    = rem / WO_;
    const int w    = rem - h * WO_;
    const int hw   = h * WO_ + w;
    const float* xb   = x   + b * (C_ * H_ * W_);
    const float* offb = off + b * (O_ * HOWO_);
    float* sv = sval + wv * VAL_STRIDE;

    // Zero-fill the pad row (147) so the GEMM LDS reads are unconditional.
    if (half == 0) sv[CK_ * 16 + p] = 0.0f;

    // ---- gather phase: build val[ck][p] tile in LDS ----
    // 32 lanes = 16 pixels x 2 kernel positions per iteration.
    int ky = half / 7, kx = half - ky * 7;     // decode of kk = half (0 or 1)
    for (int kk2 = 0; kk2 < KHW_; kk2 += 2) {
        const int kk = kk2 + half;
        if (kk < KHW_) {
            const float oy = offb[(kk * 2 + 0) * HOWO_ + hw];
            const float ox = offb[(kk * 2 + 1) * HOWO_ + hw];
            const float py = (float)(h * 2 - 3 + ky) + oy;
            const float px = (float)(w * 2 - 3 + kx) + ox;
            const float y0f = floorf(py);
            const float x0f = floorf(px);
            const float ly = py - y0f, lx = px - x0f;
            const float hy = 1.0f - ly, hx = 1.0f - lx;
            const int iy0 = (int)y0f, ix0 = (int)x0f;
            const float w00 = hy * hx, w01 = hy * lx;
            const float w10 = ly * hx, w11 = ly * lx;
            #pragma unroll
            for (int c = 0; c < C_; ++c) {
                const float* xc = xb + c * (H_*W_);
                float v = sample_x(xc, iy0,     ix0    ) * w00
                        + sample_x(xc, iy0,     ix0 + 1) * w01
                        + sample_x(xc, iy0 + 1, ix0    ) * w10
                        + sample_x(xc, iy0 + 1, ix0 + 1) * w11;
                sv[(c * KHW_ + kk) * 16 + p] = v;
            }
        }
        // kk advances by 2: kx += 2 with carry (branchless).
        kx += 2;
        if (kx >= 7) { kx -= 7; ky += 1; }
    }
    __syncthreads();   // LDS visibility (also guarantees full EXEC at WMMA)

    // ---- GEMM phase ----
    v8f acc[7];
    #pragma unroll
    for (int ot = 0; ot < 7; ++ot) acc[ot] = zero8();

    for (int ks = 0; ks < CKP_; ks += 4) {
        const int k0 = ks + 2 * half;
        v2f bfrag;                                // pad row 147 is zero in LDS
        bfrag.x = sv[(k0    ) * 16 + p];
        bfrag.y = sv[(k0 + 1) * 16 + p];
        #pragma unroll
        for (int ot = 0; ot < 7; ++ot) {
            const int o  = ot * 16 + p;
            const bool vo = (o < O_);
            const float* wrow = dw + min(o, O_ - 1) * CK_;
            v2f afrag;
            afrag.x = wgt_val(wrow, vo, k0);
            afrag.y = wgt_val(wrow, vo, k0 + 1);
            acc[ot] = __builtin_amdgcn_wmma_f32_16x16x4_f32(
                false, afrag, false, bfrag, (short)0, acc[ot], false, false);
        }
    }

    #pragma unroll
    for (int ot = 0; ot < 7; ++ot) {
        #pragma unroll
        for (int v = 0; v < 8; ++v) {
            const int o = ot * 16 + v + 8 * half;
            if (o < O_) out[(b * O_ + o) * HOWO_ + hw] = acc[ot][v];
        }
    }
}

extern "C" void kernel_launch(void* const* d_in, const int* in_sizes, int n_in,
                              void* d_out, int out_size, void* d_ws, size_t ws_size,
                              hipStream_t stream) {
    const float* x  = (const float*)d_in[0];   // (16,3,224,224)
    const float* ow = (const float*)d_in[1];   // (98,3,7,7)
    const float* dw = (const float*)d_in[2];   // (98,3,7,7)
    float* out = (float*)d_out;                // (16,98,112,112)
    float* off = (float*)d_ws;                 // offset intermediate, 78.7 MB

    const int tiles  = NPIX_ / 16;                    // 12544
    const int blocks = tiles / WAVES_PER_BLOCK;       // 3136
    dim3 block(32 * WAVES_PER_BLOCK);

    offset_conv_kernel<<<blocks, block, 0, stream>>>(x, ow, off);
    deform_conv_kernel<<<blocks, block, 0, stream>>>(x, dw, off, out);
}